// Encoder_91319594647569
// MI455X (gfx1250) — compile-verified
//
#include <hip/hip_runtime.h>

// GCN encoder for MI455X (gfx1250, wave32).
// d_in order: x[N*128], edge_index[2*E] (int32), W1[128*64], b1[64],
//             W_mu[64*64], b_mu[64], W_ls[64*64], b_ls[64]
// d_out: mu[N*64] ++ logstd[N*64] (fp32)
// d_ws layout (floats): dinv[N (padded to 100096)], bufA[N*64], bufB[N*64]

typedef __attribute__((ext_vector_type(2))) float v2f;
typedef __attribute__((ext_vector_type(8))) float v8f;
typedef __attribute__((ext_vector_type(4))) int   v4i;

#define IN_C 128
#define HID  64

// ---------------------------------------------------------------------------
// Native CDNA5 no-return f32 atomic add (GLOBAL_ATOMIC_ADD_F32, GV mode).
// Relaxed / device scope matches segment_sum's unordered float reduction.
// ---------------------------------------------------------------------------
__device__ __forceinline__ void atomic_add_f32(float* p, float v) {
  asm volatile("global_atomic_add_f32 %0, %1, off" : : "v"(p), "v"(v) : "memory");
}

// ---------------------------------------------------------------------------
// 16B global -> LDS async copy (GLOBAL_LOAD_ASYNC_TO_LDS_B128, ASYNCcnt).
// Use the clang builtin so the LDS write is visible to the optimizer (an
// asm-only version got the subsequent ds_loads folded away). Diagnostic-
// derived signature: (v4i* src, addrspace(3) v4i* dst, imm offset, imm cpol).
// ---------------------------------------------------------------------------
#if defined(__has_builtin) && __has_builtin(__builtin_amdgcn_global_load_async_to_lds_b128)
#define HAVE_ASYNC_LDS 1
typedef __attribute__((address_space(3))) v4i LV4;
__device__ __forceinline__ void async_copy16(const void* gsrc, void* ldst) {
  __builtin_amdgcn_global_load_async_to_lds_b128((v4i*)gsrc, (LV4*)ldst, 0, 0);
}
#else
#define HAVE_ASYNC_LDS 0
__device__ __forceinline__ void async_copy16(const void* gsrc, void* ldst) {
  *(float4*)ldst = *(const float4*)gsrc;  // compiler-visible fallback
}
#endif

__device__ __forceinline__ void wait_async_all() {
#if HAVE_ASYNC_LDS
  asm volatile("s_wait_asynccnt 0x0" : : : "memory");
#endif
}

// ---------------------------------------------------------------------------
// Degree accumulation: one thread per edge, atomic +1.0 into deg[dst].
// ---------------------------------------------------------------------------
__global__ __launch_bounds__(256) void deg_kernel(const int* __restrict__ dst,
                                                  float* __restrict__ deg, int E) {
  int e = blockIdx.x * 256 + threadIdx.x;
  if (e < E) atomic_add_f32(&deg[dst[e]], 1.0f);
}

// dinv[i] = rsqrt(deg[i] + 1)   (in place over the deg buffer)
__global__ __launch_bounds__(256) void dinv_kernel(float* __restrict__ deg, int n) {
  int i = blockIdx.x * 256 + threadIdx.x;
  if (i < n) deg[i] = __frsqrt_rn(deg[i] + 1.0f);
}

// ---------------------------------------------------------------------------
// Full-precision WMMA GEMM:  C[N,64] = X[N,K] @ W[K,64],  K in {128, 64}.
// Block = 256 threads = 8 wave32s; each block computes a 32x64 tile:
//   wave w -> 16x16 tile at rows (2*bx + w/4)*16, cols (w%4)*16.
// W staged to LDS via async-to-LDS DMA. Uses V_WMMA_F32_16X16X4_F32.
// A frag (16x4 f32): lanes 0-15 hold K=k,k+1 ; lanes 16-31 hold K=k+2,k+3.
// B frag (4x16 f32): v0/v1 mirror the A split across the lane halves.
// C/D (16x16 f32): VGPR g, lanes 0-15 -> M=g, lanes 16-31 -> M=g+8.
// Grid is exact (N % 32 == 0) so EXEC is all ones at every WMMA.
// ---------------------------------------------------------------------------
__global__ __launch_bounds__(256) void gemm_wmma_f32(const float* __restrict__ X,
                                                     const float* __restrict__ W,
                                                     float* __restrict__ C, int K) {
  __shared__ float Ws[IN_C * 64];  // up to 32 KB of the 320 KB/WGP LDS

  const int tid = threadIdx.x;
  // Stage W[K,64] into LDS: K*16 float4s, 256 lanes -> 4 or 8 DMAs each.
  const int nW4 = K * 16;
  const float4* W4 = (const float4*)W;
  float4* Ws4 = (float4*)Ws;
  for (int i = tid; i < nW4; i += 256)
    async_copy16(W4 + i, Ws4 + i);
  wait_async_all();
  __syncthreads();

  const int wave = tid >> 5;   // 0..7
  const int lane = tid & 31;
  const int ln   = lane & 15;  // row within tile (A) / col within tile (B,C)
  const int hi   = lane >> 4;  // 0: K pair {0,1}, 1: K pair {2,3}
  const int m0   = (blockIdx.x * 2 + (wave >> 2)) * 16;
  const int n0   = (wave & 3) * 16;

  v8f acc = {};
  const float* xrow = X + (size_t)(m0 + ln) * K + (hi << 1);
  for (int k = 0; k < K; k += 4) {
    v2f a = *(const v2f*)(xrow + k);                       // global_load_b64
    const float* wp = Ws + (k + (hi << 1)) * 64 + n0 + ln; // LDS
    v2f b;
    b.x = wp[0];
    b.y = wp[64];
    acc = __builtin_amdgcn_wmma_f32_16x16x4_f32(
        /*neg_a=*/false, a, /*neg_b=*/false, b,
        /*c_mod=*/(short)0, acc, /*reuse_a=*/false, /*reuse_b=*/false);
  }

  // Store D: lane half 'hi' owns rows m0 + 8*hi + g, column n0 + ln.
  float* crow = C + (size_t)(m0 + hi * 8) * 64 + n0 + ln;
#pragma unroll
  for (int g = 0; g < 8; ++g) crow[(size_t)g * 64] = acc[g];
}

// ---------------------------------------------------------------------------
// out[i,f] = pre[i,f] * dinv[i]^2 + bias[f]   (self-loop term + bias; also
// initializes the scatter accumulation target)
// ---------------------------------------------------------------------------
__global__ __launch_bounds__(256) void init_out_kernel(float* __restrict__ out,
                                                       const float* __restrict__ pre,
                                                       const float* __restrict__ dinv,
                                                       const float* __restrict__ bias,
                                                       int total) {
  int idx = blockIdx.x * 256 + threadIdx.x;
  if (idx < total) {
    int i = idx >> 6;
    int f = idx & 63;
    float d = dinv[i];
    out[idx] = pre[idx] * d * d + bias[f];
  }
}

// ---------------------------------------------------------------------------
// Edge scatter: one wave per edge. Each lane handles 2 of the 64 features
// (float2 gather from pre[src], 2 native f32 atomic adds into out[dst]).
// src/dst/norm are wave-uniform -> scalarized by the compiler.
// ---------------------------------------------------------------------------
__global__ __launch_bounds__(256) void scatter_kernel(float* __restrict__ out,
                                                      const float* __restrict__ pre,
                                                      const int* __restrict__ src,
                                                      const int* __restrict__ dst,
                                                      const float* __restrict__ dinv,
                                                      int E) {
  int e    = (blockIdx.x << 3) + (threadIdx.x >> 5);
  int lane = threadIdx.x & 31;
  if (e >= E) return;
  int s = src[e];
  int d = dst[e];
  float norm = dinv[s] * dinv[d];
  const float2 v = *(const float2*)(pre + (size_t)s * 64 + lane * 2);
  float* o = out + (size_t)d * 64 + lane * 2;
  atomic_add_f32(o, norm * v.x);
  atomic_add_f32(o + 1, norm * v.y);
}

__global__ __launch_bounds__(256) void relu_kernel(float* __restrict__ x, int n) {
  int i = blockIdx.x * 256 + threadIdx.x;
  if (i < n) x[i] = fmaxf(x[i], 0.0f);
}

// ---------------------------------------------------------------------------
extern "C" void kernel_launch(void* const* d_in, const int* in_sizes, int n_in,
                              void* d_out, int out_size, void* d_ws, size_t ws_size,
                              hipStream_t stream) {
  const float* x   = (const float*)d_in[0];
  const int*   ei  = (const int*)d_in[1];
  const float* W1  = (const float*)d_in[2];
  const float* b1  = (const float*)d_in[3];
  const float* Wmu = (const float*)d_in[4];
  const float* bmu = (const float*)d_in[5];
  const float* Wls = (const float*)d_in[6];
  const float* bls = (const float*)d_in[7];

  const int N = in_sizes[0] / IN_C;   // 100000
  const int E = in_sizes[1] / 2;      // 1600000
  const int* src = ei;
  const int* dst = ei + E;

  float* dinv = (float*)d_ws;
  float* bufA = dinv + 100096;              // pre-activation GEMM output
  float* bufB = bufA + (size_t)N * HID;     // hidden features h
  float* mu   = (float*)d_out;
  float* ls   = mu + (size_t)N * HID;

  const int NF        = N * HID;            // 6.4M feature elements
  const int nf_blocks = (NF + 255) / 256;
  const int n_blocks  = (N + 255) / 256;
  const int e_blocks  = (E + 255) / 256;
  const int ew_blocks = (E + 7) / 8;        // wave-per-edge
  const int gemm_blks = N / 32;             // 3125, exact

  // --- degrees & normalization ---
  (void)hipMemsetAsync(dinv, 0, (size_t)N * sizeof(float), stream);
  deg_kernel<<<e_blocks, 256, 0, stream>>>(dst, dinv, E);
  dinv_kernel<<<n_blocks, 256, 0, stream>>>(dinv, N);

  // --- layer 1: h = relu(norm-aggregate(x @ W1) + b1) ---
  gemm_wmma_f32<<<gemm_blks, 256, 0, stream>>>(x, W1, bufA, IN_C);
  init_out_kernel<<<nf_blocks, 256, 0, stream>>>(bufB, bufA, dinv, b1, NF);
  scatter_kernel<<<ew_blocks, 256, 0, stream>>>(bufB, bufA, src, dst, dinv, E);
  relu_kernel<<<nf_blocks, 256, 0, stream>>>(bufB, NF);

  // --- mu head ---
  gemm_wmma_f32<<<gemm_blks, 256, 0, stream>>>(bufB, Wmu, bufA, HID);
  init_out_kernel<<<nf_blocks, 256, 0, stream>>>(mu, bufA, dinv, bmu, NF);
  scatter_kernel<<<ew_blocks, 256, 0, stream>>>(mu, bufA, src, dst, dinv, E);

  // --- logstd head ---
  gemm_wmma_f32<<<gemm_blks, 256, 0, stream>>>(bufB, Wls, bufA, HID);
  init_out_kernel<<<nf_blocks, 256, 0, stream>>>(ls, bufA, dinv, bls, NF);
  scatter_kernel<<<ew_blocks, 256, 0, stream>>>(ls, bufA, src, dst, dinv, E);
}